// HierarchicalMultilabelLoss_24721831755867
// MI455X (gfx1250) — compile-verified
//
#include <hip/hip_runtime.h>
#include <hip/hip_bf16.h>

// Problem constants (from reference)
#define B_     256
#define G_     16
#define L_     16
#define C_     8192
#define K_     8
#define NODES  (B_ * L_)        // 4096
#define SPLIT  2                // blocks per sample (C split in halves)
#define HALF_C (C_ / SPLIT)     // 4096 floats = 16 KB chunk

typedef unsigned int u32;
typedef u32 u32x4 __attribute__((ext_vector_type(4)));
typedef int  i32x4 __attribute__((ext_vector_type(4)));
typedef int  i32x8 __attribute__((ext_vector_type(8)));

// ---------------------------------------------------------------------------
// Fast math helpers (v_exp_f32 / v_log_f32 / v_rcp_f32 paths)
// ---------------------------------------------------------------------------
__device__ __forceinline__ float fsigmoid(float x) {
    float e = __expf(-x);
    return __builtin_amdgcn_rcpf(1.0f + e);
}

// t=0 loss given s=sigmoid(x): xs_neg=min(1.05-s,1); -log(max(xs_neg,eps))*(1-xs_neg)^4
// with 1-xs_neg = max(s-0.05, 0).
__device__ __forceinline__ float f_neg(float s) {
    float xn = fminf(1.05f - s, 1.0f);
    float q  = fmaxf(s - 0.05f, 0.0f);
    float q2 = q * q;
    return -__logf(fmaxf(xn, 1e-8f)) * (q2 * q2);
}

// t=1 loss: -log(max(s,eps)) * (1-s)
__device__ __forceinline__ float f_pos(float s) {
    return -__logf(fmaxf(s, 1e-8f)) * (1.0f - s);
}

// ---------------------------------------------------------------------------
// TDM: DMA `nf` contiguous floats from global into LDS (1-D tile descriptor,
// D# layout per CDNA5 ISA sec 8.3 / 8.4). Issued by one wave; completion
// tracked with TENSORcnt.
// ---------------------------------------------------------------------------
#if defined(__gfx1250__) && __has_builtin(__builtin_amdgcn_tensor_load_to_lds)
#define HAVE_TDM 1
#endif

__device__ __forceinline__ void tdm_issue(const float* gsrc, float* lds_dst, u32 nf) {
#ifdef HAVE_TDM
    u32 lds_off = (u32)(unsigned long long)(void*)lds_dst;  // low 32b of flat LDS ptr = LDS offset
    unsigned long long ga = (unsigned long long)gsrc;

    u32x4 g0;
    g0.x = 1u;                                            // count=1 (valid D#)
    g0.y = lds_off;                                       // lds_addr (bytes)
    g0.z = (u32)(ga & 0xFFFFFFFFu);                       // global_addr[31:0]
    g0.w = (u32)((ga >> 32) & 0x01FFFFFFu) | (2u << 30);  // addr[56:32] | type=2

    i32x8 g1;
    g1[0] = (int)(2u << 16);                // data_size=2 (4B); no mask/pad/iterate
    g1[1] = (int)((nf & 0xFFFFu) << 16);    // tensor_dim0 low16
    g1[2] = (int)((1u << 16) | (nf >> 16)); // tensor_dim0 hi16 | tensor_dim1=1
    g1[3] = (int)(nf << 16);                // tile_dim0 = nf
    g1[4] = 0;                              // tile_dim1=0, tile_dim2=0
    g1[5] = (int)nf;                        // tensor_dim0_stride low32
    g1[6] = (int)((nf & 0xFFFFu) << 16);    // stride0 hi16=0 | stride1 lo16
    g1[7] = (int)(nf >> 16);                // stride1 hi32

    i32x4 z4 = {0, 0, 0, 0};
#if __clang_major__ >= 23
    i32x8 z8 = {0, 0, 0, 0, 0, 0, 0, 0};
    __builtin_amdgcn_tensor_load_to_lds(g0, g1, z4, z4, z8, 0);
#else
    __builtin_amdgcn_tensor_load_to_lds(g0, g1, z4, z4, 0);
#endif
#else
    (void)gsrc; (void)lds_dst; (void)nf;
#endif
}

__device__ __forceinline__ void tdm_wait(int pending) {
#ifdef HAVE_TDM
#if __has_builtin(__builtin_amdgcn_s_wait_tensorcnt)
    if (pending) __builtin_amdgcn_s_wait_tensorcnt(1);
    else         __builtin_amdgcn_s_wait_tensorcnt(0);
#else
    if (pending) asm volatile("s_wait_tensorcnt 0x1" ::: "memory");
    else         asm volatile("s_wait_tensorcnt 0x0" ::: "memory");
#endif
#else
    (void)pending;
#endif
}

// ---------------------------------------------------------------------------
// Kernel 1: grid = B*SPLIT blocks of 1024 threads. Block (b, hpart) handles
// columns [hpart*HALF_C, (hpart+1)*HALF_C) for all 16 levels of sample b.
// Dedups the 16 group params, TDM-streams each distinct row-half through a
// double-buffered 16KB LDS stage, computes the shared all-negative base sum
// once per distinct row, applies per-level label corrections in-flight.
// ---------------------------------------------------------------------------
__global__ __launch_bounds__(1024)
void hml_node_kernel(const float* __restrict__ logits,
                     const int*  __restrict__ param,
                     const int*  __restrict__ label,
                     const int*  __restrict__ label_num,
                     float* __restrict__ ws) {
    __shared__ __align__(16) float buf[2][HALF_C];  // 32 KB double buffer
    __shared__ float wred[32];                      // per-wave partials
    __shared__ float s_base;
    __shared__ int   s_g[L_];                       // distinct group ids
    __shared__ int   s_map[L_];                     // level -> distinct idx (-1 invalid)
    __shared__ int   s_n;

    const int b     = blockIdx.x >> 1;
    const int hpart = blockIdx.x & 1;
    const int tid   = threadIdx.x;
    const int lane  = tid & 31;
    const int wid   = tid >> 5;

    // --- Setup: dedup the 16 group params for this sample (serial, tiny) ---
    if (tid == 0) {
        int n = 0;
        for (int l = 0; l < L_; ++l) {
            int pv = param[b * L_ + l];
            if (pv < 0) { s_map[l] = -1; continue; }
            int p = pv > (G_ - 1) ? (G_ - 1) : pv;
            int idx = -1;
            for (int m = 0; m < n; ++m) if (s_g[m] == p) { idx = m; break; }
            if (idx < 0) { idx = n; s_g[n++] = p; }
            s_map[l] = idx;
        }
        s_n = n;
    }
    __syncthreads();

    const int n = s_n;

    // Invalid levels contribute zero (jnp.where(valid, ..., 0)).
    if (tid < L_ && s_map[tid] < 0)
        ws[(b * L_ + tid) * SPLIT + hpart] = 0.0f;

    // --- Prologue: issue first DMA (wave 0 only; TDM ignores EXEC, branch is
    // wave-uniform so only wave 0's stream contains the instruction) ---
    if (n > 0 && tid < 32) {
        int g = s_g[0];
        tdm_issue(logits + ((size_t)g * B_ + (size_t)b) * C_ + (size_t)hpart * HALF_C,
                  buf[0], HALF_C);
    }

    for (int i = 0; i < n; ++i) {
        if (tid < 32) {
            if (i + 1 < n) {
                int g = s_g[i + 1];
                tdm_issue(logits + ((size_t)g * B_ + (size_t)b) * C_ + (size_t)hpart * HALF_C,
                          buf[(i + 1) & 1], HALF_C);
            }
            tdm_wait(i + 1 < n);      // oldest DMA (row i) complete
        }
        __syncthreads();

        const float* cur = buf[i & 1];

        // Base pass: every element as negative target (4 elems/thread).
        float4 v = ((const float4*)cur)[tid];
        float acc = f_neg(fsigmoid(v.x)) + f_neg(fsigmoid(v.y)) +
                    f_neg(fsigmoid(v.z)) + f_neg(fsigmoid(v.w));

        // Corrections: level-thread l swaps f_neg -> f_pos at its distinct,
        // valid label positions that fall inside this C-half.
        float corr = 0.0f;
        if (tid < L_ && s_map[tid] == i) {
            const int node = b * L_ + tid;
            int num = label_num[node];
            if (num < 0) num = 0;
            if (num > K_) num = K_;
            int labs[K_];
            for (int k = 0; k < K_; ++k)
                labs[k] = (k < num) ? label[node * K_ + k] : -1;
            for (int k = 0; k < num; ++k) {
                int c = labs[k];
                bool dup = false;
                for (int m = 0; m < k; ++m) dup |= (labs[m] == c);
                if (dup) continue;
                if (c < 0) c = 0;
                if (c >= C_) c = C_ - 1;
                if (c >= hpart * HALF_C && c < (hpart + 1) * HALF_C) {
                    float s = fsigmoid(cur[c - hpart * HALF_C]);
                    corr += f_pos(s) - f_neg(s);
                }
            }
        }

        // Deterministic block reduction: wave32 butterfly + 32-entry stage.
        for (int off = 16; off > 0; off >>= 1)
            acc += __shfl_xor(acc, off, 32);
        if (lane == 0) wred[wid] = acc;
        __syncthreads();
        if (tid < 32) {
            float t = wred[tid];
            for (int off = 16; off > 0; off >>= 1)
                t += __shfl_xor(t, off, 32);
            if (tid == 0) s_base = t;
        }
        __syncthreads();

        if (tid < L_ && s_map[tid] == i)
            ws[(b * L_ + tid) * SPLIT + hpart] = s_base + corr;
        // No trailing barrier needed: all buf[] reads completed before the
        // first reduction barrier above, so next iteration's DMA issue into
        // this same buffer cannot race with them.
    }

#ifndef HAVE_TDM
    // Fallback path (no TDM builtin): empty placeholder, computes nothing.
    for (int i = 0; i < n; ++i) {
        // (kept minimal; TDM path is the expected configuration)
    }
#endif
}

// ---------------------------------------------------------------------------
// Kernel 2: deterministic reduction of NODES*SPLIT partials -> mean over B.
// ---------------------------------------------------------------------------
__global__ __launch_bounds__(256)
void hml_final_reduce(const float* __restrict__ partials,
                      float* __restrict__ out) {
    __shared__ float red[256];
    const int tid = threadIdx.x;
    float a = 0.0f;
    #pragma unroll
    for (int i = 0; i < (NODES * SPLIT) / 256; ++i)
        a += partials[tid * ((NODES * SPLIT) / 256) + i];
    red[tid] = a;
    __syncthreads();
    for (int s = 128; s > 0; s >>= 1) {
        if (tid < s) red[tid] += red[tid + s];
        __syncthreads();
    }
    if (tid == 0) out[0] = red[0] * (1.0f / (float)B_);
}

// ---------------------------------------------------------------------------
extern "C" void kernel_launch(void* const* d_in, const int* in_sizes, int n_in,
                              void* d_out, int out_size, void* d_ws, size_t ws_size,
                              hipStream_t stream) {
    const float* logits    = (const float*)d_in[0];  // (G,B,C) f32
    const int*   param     = (const int*)d_in[1];    // (B,L)   i32
    const int*   label     = (const int*)d_in[2];    // (B,L,K) i32
    const int*   label_num = (const int*)d_in[3];    // (B,L)   i32
    float* out = (float*)d_out;
    float* ws  = (float*)d_ws;                        // NODES*SPLIT floats scratch

    hml_node_kernel<<<B_ * SPLIT, 1024, 0, stream>>>(logits, param, label, label_num, ws);
    hml_final_reduce<<<1, 256, 0, stream>>>(ws, out);
}